// MoeRouter_50869592655342
// MI455X (gfx1250) — compile-verified
//
#include <hip/hip_runtime.h>
#include <math.h>

#define T_TOK 32768
#define DIM   2048
#define NEXP  64
#define TOPK  2

#define KC  64    // K-chunk staged in LDS
#define MT  32    // tokens per workgroup
#define PAD 68    // padded LDS row stride (floats): 4-bank stride, 16B aligned

typedef float v2f __attribute__((ext_vector_type(2)));
typedef float v8f __attribute__((ext_vector_type(8)));

__global__ __launch_bounds__(256) void moe_router_main(
    const float* __restrict__ x, const float* __restrict__ W,
    float* __restrict__ out, float* __restrict__ gcnt, float* __restrict__ gsum)
{
    __shared__ float xs[MT * PAD];      // x tile   32 x 64 (padded)
    __shared__ float ws[NEXP * PAD];    // W tile   64 x 64 (padded)
    __shared__ float lb[MT * NEXP];     // logits   32 x 64
    __shared__ float hcnt[NEXP];
    __shared__ float hsum[NEXP];

    const int tid   = threadIdx.x;
    const int lane  = tid & 31;
    const int wave  = tid >> 5;
    const int mhalf = wave >> 2;        // 0..1 : which 16-token half
    const int ntile = wave & 3;         // 0..3 : which 16-expert tile
    const int l16   = lane & 15;
    const int h     = lane >> 4;        // lane half selects K pair per ISA layout
    const int tokBase = blockIdx.x * MT;

    if (tid < NEXP) { hcnt[tid] = 0.f; hsum[tid] = 0.f; }

    v8f acc = {};

    for (int k0 = 0; k0 < DIM; k0 += KC) {
        // stage x tile: 32x64 floats = 512 float4, 2 per thread, coalesced
        #pragma unroll
        for (int i = 0; i < 2; ++i) {
            int id  = tid + i * 256;
            int row = id >> 4, c4 = (id & 15) << 2;
            float4 v = *reinterpret_cast<const float4*>(
                &x[(size_t)(tokBase + row) * DIM + k0 + c4]);
            *reinterpret_cast<float4*>(&xs[row * PAD + c4]) = v;
        }
        // stage W tile: 64x64 floats = 1024 float4, 4 per thread (L2-resident)
        #pragma unroll
        for (int i = 0; i < 4; ++i) {
            int id  = tid + i * 256;
            int row = id >> 4, c4 = (id & 15) << 2;
            float4 v = *reinterpret_cast<const float4*>(
                &W[(size_t)row * DIM + k0 + c4]);
            *reinterpret_cast<float4*>(&ws[row * PAD + c4]) = v;
        }
        __syncthreads();

        const float* xrow = &xs[(mhalf * 16 + l16) * PAD];
        const float* wrow = &ws[(ntile * 16 + l16) * PAD];
        #pragma unroll
        for (int kk = 0; kk < KC; kk += 4) {
            // A 16x4 f32: lanes 0-15 hold K={kk,kk+1}, lanes 16-31 K={kk+2,kk+3}
            v2f a = *reinterpret_cast<const v2f*>(&xrow[kk + 2 * h]);
            // B 4x16 f32: N = lane%16, same K striping; B[k][n] = W[n][k]
            v2f b = *reinterpret_cast<const v2f*>(&wrow[kk + 2 * h]);
            acc = __builtin_amdgcn_wmma_f32_16x16x4_f32(
                false, a, false, b, (short)0, acc, false, false);
        }
        __syncthreads();
    }

    // C/D layout: VGPR v -> M = v + 8*(lane>>4), N = lane&15
    #pragma unroll
    for (int v = 0; v < 8; ++v) {
        int row = mhalf * 16 + h * 8 + v;
        int col = ntile * 16 + l16;
        lb[row * NEXP + col] = acc[v];
    }
    __syncthreads();

    if (tid < MT) {
        const float* lrow = &lb[tid * NEXP];
        float m = lrow[0];
        for (int e = 1; e < NEXP; ++e) m = fmaxf(m, lrow[e]);
        float ssum = 0.f;
        float b1 = -INFINITY, b2 = -INFINITY; int i1 = 0, i2 = 0;
        for (int e = 0; e < NEXP; ++e) {
            float le = lrow[e];
            ssum += __expf(le - m);
            if (le > b1)      { b2 = b1; i2 = i1; b1 = le; i1 = e; }
            else if (le > b2) { b2 = le; i2 = e; }
        }
        float inv = 1.f / ssum;
        float p1 = __expf(b1 - m) * inv;
        float p2 = __expf(b2 - m) * inv;
        float wn = 1.f / (p1 + p2 + 1e-9f);
        int t = tokBase + tid;
        out[(size_t)t * 2 + 0] = p1 * wn;
        out[(size_t)t * 2 + 1] = p2 * wn;
        out[(size_t)T_TOK * 2 + (size_t)t * 2 + 0] = (float)i1;
        out[(size_t)T_TOK * 2 + (size_t)t * 2 + 1] = (float)i2;
        atomicAdd(&hcnt[i1], 1.f); atomicAdd(&hcnt[i2], 1.f);
        atomicAdd(&hsum[i1], p1);  atomicAdd(&hsum[i2], p2);
    }
    __syncthreads();
    if (tid < NEXP) {
        atomicAdd(&gcnt[tid], hcnt[tid]);
        atomicAdd(&gsum[tid], hsum[tid]);
    }
}

__global__ void moe_router_zero(float* g) { g[threadIdx.x] = 0.f; }

__global__ void moe_router_aux(const float* __restrict__ gcnt,
                               const float* __restrict__ gsum,
                               float* __restrict__ out)
{
    __shared__ float red[NEXP];
    int t = threadIdx.x;
    float cnt  = gcnt[t], sm = gsum[t];
    float frac = cnt / ((float)(T_TOK * TOPK) + 1e-6f);
    float avg  = sm / fmaxf(cnt, 1e-6f);
    red[t] = frac * avg;
    __syncthreads();
    for (int s = 32; s > 0; s >>= 1) {
        if (t < s) red[t] += red[t + s];
        __syncthreads();
    }
    if (t == 0) out[(size_t)T_TOK * TOPK * 2] = red[0] * (float)NEXP;
}

extern "C" void kernel_launch(void* const* d_in, const int* in_sizes, int n_in,
                              void* d_out, int out_size, void* d_ws, size_t ws_size,
                              hipStream_t stream) {
    const float* x = (const float*)d_in[0];   // [T, D] fp32
    const float* W = (const float*)d_in[1];   // [E, D] fp32
    float* out  = (float*)d_out;              // top_w | top_idx | aux
    float* gcnt = (float*)d_ws;               // [64]
    float* gsum = gcnt + NEXP;                // [64]

    moe_router_zero<<<1, 2 * NEXP, 0, stream>>>(gcnt);
    moe_router_main<<<T_TOK / MT, 256, 0, stream>>>(x, W, out, gcnt, gsum);
    moe_router_aux<<<1, NEXP, 0, stream>>>(gcnt, gsum, out);
}